// EdgeNodeCountingLayer_90443421319693
// MI455X (gfx1250) — compile-verified
//
#include <hip/hip_runtime.h>
#include <stdint.h>

typedef uint32_t u32;
typedef __attribute__((ext_vector_type(4))) u32 u32x4;
typedef __attribute__((ext_vector_type(8))) u32 u32x8;

#define B_DIM 64
#define IN_DIM 512
#define OUT_DIM 512
#define OTILE 8            // output nodes per block (== waves per block)

// ---------------------------------------------------------------------------
// Threefry-2x32 (20 rounds), matching jax._src.prng.threefry2x32
// ---------------------------------------------------------------------------
__device__ __forceinline__ u32 rotl32(u32 x, u32 r) { return (x << r) | (x >> (32u - r)); }

__device__ __forceinline__ void tf2x32(u32 k0, u32 k1, u32 c0, u32 c1, u32& o0, u32& o1) {
  u32 ks0 = k0, ks1 = k1, ks2 = 0x1BD11BDAu ^ k0 ^ k1;
  u32 x0 = c0 + ks0, x1 = c1 + ks1;
#define TFR(r) x0 += x1; x1 = rotl32(x1, r); x1 ^= x0;
  TFR(13) TFR(15) TFR(26) TFR(6)   x0 += ks1; x1 += ks2 + 1u;
  TFR(17) TFR(29) TFR(16) TFR(24)  x0 += ks2; x1 += ks0 + 2u;
  TFR(13) TFR(15) TFR(26) TFR(6)   x0 += ks0; x1 += ks1 + 3u;
  TFR(17) TFR(29) TFR(16) TFR(24)  x0 += ks1; x1 += ks2 + 4u;
  TFR(13) TFR(15) TFR(26) TFR(6)   x0 += ks2; x1 += ks0 + 5u;
#undef TFR
  o0 = x0; o1 = x1;
}

// bits[e] of threefry_2x32(key, iota(2*half)): counts split into halves,
// pair q processed as (q, q+half); out = concat(out0s, out1s).
__device__ __forceinline__ u32 tf_elem(u32 k0, u32 k1, u32 e, u32 half) {
  u32 q = (e < half) ? e : (e - half);
  u32 o0, o1;
  tf2x32(k0, k1, q, q + half, o0, o1);
  return (e < half) ? o0 : o1;
}

// jax.random.uniform(minval=tiny, maxval=1) -> gumbel
__device__ __forceinline__ float gumbel_from_bits(u32 bits) {
  const float tiny = 1.17549435e-38f;
  float f = __uint_as_float((bits >> 9) | 0x3f800000u) - 1.0f;   // [0,1)
  float u = f * (1.0f - tiny) + tiny;
  u = fmaxf(u, tiny);
  return -logf(-logf(u));
}

// ---------------------------------------------------------------------------
// Kernel 1: reproduce gumbel-softmax(hard) selections.
//   op_idx[o]   : argmax_j (otc[o,j] + g_op[o,j])           (512 ints)
//   mask[o*16+k]: bit t set iff edge (type 0) selected for i = 32k + t
// Thread g handles (o = g/16, mask word k = g%16): 64 threefry evals.
// ---------------------------------------------------------------------------
__global__ __launch_bounds__(256) void fuzzy_select_kernel(
    const float* __restrict__ etc,   // [512,2,512,2] edge_type_count (logits)
    const float* __restrict__ otc,   // [512,2]       operator_type_count
    int* __restrict__ op_idx, u32* __restrict__ mask) {
  u32 g = blockIdx.x * 256u + threadIdx.x;
  if (g >= (u32)OUT_DIM * 16u) return;
  u32 o = g >> 4, k = g & 15u;

  // key(42) = (0,42); split -> k_op=(p0.a,p1.a), k_edge=(p0.b,p1.b)
  u32 p0a, p0b, p1a, p1b;
  tf2x32(0u, 42u, 0u, 2u, p0a, p0b);
  tf2x32(0u, 42u, 1u, 3u, p1a, p1b);
  const u32 ko0 = p0a, ko1 = p1a;   // k_op
  const u32 ke0 = p0b, ke1 = p1b;   // k_edge

  // operator selection: shape (512,2), size 1024, half = 512
  float s0 = otc[2u * o]      + gumbel_from_bits(tf_elem(ko0, ko1, 2u * o,      512u));
  float s1 = otc[2u * o + 1u] + gumbel_from_bits(tf_elem(ko0, ko1, 2u * o + 1u, 512u));
  u32 kop = (s0 >= s1) ? 0u : 1u;   // argmax, first occurrence on ties
  if (k == 0u) op_idx[o] = (int)kop;

  // edge selection: shape (512,2,512,2), size 1048576, half = 524288
  const u32 half = 524288u;
  u32 word = 0u;
  u32 base = (2u * o + kop) * 512u;
  for (u32 t = 0u; t < 32u; ++t) {
    u32 i = k * 32u + t;
    u32 e = (base + i) * 2u;
    float e0 = etc[e]      + gumbel_from_bits(tf_elem(ke0, ke1, e,      half));
    float e1 = etc[e + 1u] + gumbel_from_bits(tf_elem(ke0, ke1, e + 1u, half));
    if (e0 >= e1) word |= (1u << t);   // edge type 0 ("edge": value = x) selected
  }
  mask[o * 16u + k] = word;
}

// ---------------------------------------------------------------------------
// Kernel 2: main reduction. One block per OTILE=8 output nodes; one wave per
// node; lanes stripe the 512 inputs (stride-32 -> conflict-free LDS banks).
// x [64x512 f32, 128KB] is staged into LDS with one TDM descriptor
// (tensor_load_to_lds) and waited with s_wait_tensorcnt.
// ---------------------------------------------------------------------------
__global__ __launch_bounds__(256) void fuzzy_main_kernel(
    const float* __restrict__ x, const int* __restrict__ op_idx,
    const u32* __restrict__ mask, float* __restrict__ out) {
  extern __shared__ char smem[];                       // dynamic LDS
  float* xs = (float*)smem;                            // [64][512] = 131072 B
  u32*   ms = (u32*)(smem + B_DIM * IN_DIM * 4);       // [OTILE*16]
  int*   ops = (int*)(smem + B_DIM * IN_DIM * 4 + OTILE * 16 * 4);

  const int tid = threadIdx.x;
  const int o0 = blockIdx.x * OTILE;

  if (tid < OTILE * 16) ms[tid] = mask[o0 * 16 + tid];
  if (tid < OTILE)      ops[tid] = op_idx[o0 + tid];

  if (tid < 32) {  // wave 0 issues the TDM load (EXEC ignored by TDM)
    u32 ldsa = __builtin_amdgcn_groupstaticsize();     // dynamic LDS starts after static
    uint64_t ga = (uint64_t)(uintptr_t)x;
    u32x4 g0;
    g0.x = 1u;                                         // count=1, user mode, no gather
    g0.y = ldsa;                                       // lds_addr
    g0.z = (u32)ga;                                    // global_addr[31:0]
    g0.w = (u32)((ga >> 32) & 0x1FFFFFFu) | (2u << 30);// global_addr[56:32] | type=2
    u32x8 g1;
    g1.s0 = (2u << 16);            // wg_mask=0, data_size=2 (4B), no flags
    g1.s1 = (512u << 16);          // tensor_dim0 = 512 (low16 @ bits 63:48)
    g1.s2 = (64u << 16);           // tensor_dim0 hi=0 | tensor_dim1 = 64
    g1.s3 = (512u << 16);          // tensor_dim1 hi=0 | tile_dim0 = 512
    g1.s4 = 64u;                   // tile_dim1 = 64, tile_dim2 = 0
    g1.s5 = 512u;                  // tensor_dim0_stride = 512 (low32)
    g1.s6 = 0u;                    // stride0 hi16 | tensor_dim1_stride lo16 (unused)
    g1.s7 = 0u;
    asm volatile("tensor_load_to_lds %0, %1" :: "s"(g0), "s"(g1) : "memory");
    __builtin_amdgcn_s_wait_tensorcnt(0);
  }
  __syncthreads();

  const int w = tid >> 5, lane = tid & 31;
  const int o = o0 + w;
  const int op = ops[w];
  const float off = (op == 0) ? 1.0f : 0.0f;

  // per-lane selection pattern: bit k <-> input i = 32k + lane
  u32 lanebits = 0u;
#pragma unroll
  for (int k = 0; k < 16; ++k)
    lanebits |= ((ms[w * 16 + k] >> lane) & 1u) << k;

  if (op == 0) {  // t-norm: min over inputs
    for (int b = 0; b < B_DIM; ++b) {
      float acc = 1e30f;
#pragma unroll
      for (int k = 0; k < 16; ++k) {
        float xv = xs[b * IN_DIM + k * 32 + lane];
        float v = ((lanebits >> k) & 1u) ? xv : off;
        acc = fminf(acc, v);
      }
#pragma unroll
      for (int s = 16; s >= 1; s >>= 1)
        acc = fminf(acc, __shfl_xor(acc, s, 32));
      if (lane == 0) out[b * OUT_DIM + o] = acc;
    }
  } else {        // t-conorm: max over inputs
    for (int b = 0; b < B_DIM; ++b) {
      float acc = -1e30f;
#pragma unroll
      for (int k = 0; k < 16; ++k) {
        float xv = xs[b * IN_DIM + k * 32 + lane];
        float v = ((lanebits >> k) & 1u) ? xv : off;
        acc = fmaxf(acc, v);
      }
#pragma unroll
      for (int s = 16; s >= 1; s >>= 1)
        acc = fmaxf(acc, __shfl_xor(acc, s, 32));
      if (lane == 0) out[b * OUT_DIM + o] = acc;
    }
  }
}

// ---------------------------------------------------------------------------
extern "C" void kernel_launch(void* const* d_in, const int* in_sizes, int n_in,
                              void* d_out, int out_size, void* d_ws, size_t ws_size,
                              hipStream_t stream) {
  const float* x   = (const float*)d_in[0];   // [64, 512]
  const float* etc = (const float*)d_in[1];   // [512, 2, 512, 2]
  const float* otc = (const float*)d_in[2];   // [512, 2]
  float* out = (float*)d_out;                 // [64, 512]

  int* op_idx = (int*)d_ws;                            // 512 ints
  u32* mask   = (u32*)((char*)d_ws + 512 * sizeof(int)); // 512*16 u32 (32KB)

  fuzzy_select_kernel<<<32, 256, 0, stream>>>(etc, otc, op_idx, mask);

  size_t shmem = (size_t)B_DIM * IN_DIM * 4 + OTILE * 16 * 4 + OTILE * 4;
  fuzzy_main_kernel<<<OUT_DIM / OTILE, 256, shmem, stream>>>(x, op_idx, mask, out);
}